// PottsAttractor_13185549599093
// MI455X (gfx1250) — compile-verified
//
#include <hip/hip_runtime.h>
#include <cstdint>
#include <cstddef>

typedef float v2f __attribute__((ext_vector_type(2)));
typedef float v4f __attribute__((ext_vector_type(4)));
typedef float v8f __attribute__((ext_vector_type(8)));
typedef unsigned int u32x4 __attribute__((ext_vector_type(4)));
typedef int i32x4 __attribute__((ext_vector_type(4)));
typedef int i32x8 __attribute__((ext_vector_type(8)));

#define Hh 1024
#define Mm 8
#define HM 8192
static constexpr size_t HM2 = (size_t)HM * (size_t)HM;

// d_out element offsets (tuple return, concatenated flat in return order)
static constexpr size_t OFF_O   = 0;
static constexpr size_t OFF_S   = 8192;
static constexpr size_t OFF_A   = 16384;
static constexpr size_t OFF_ZI  = 24576;
static constexpr size_t OFF_ZJ  = 32768;
static constexpr size_t OFF_P   = 40960;   // scalar
static constexpr size_t OFF_PI  = 40961;
static constexpr size_t OFF_PJ  = 49153;
static constexpr size_t OFF_PIJ = 57345;
static constexpr size_t OFF_W   = OFF_PIJ + HM2;
static constexpr size_t OFF_B   = OFF_W + HM2;

__device__ __forceinline__ float log_eps(float v) {
    // log(where(x<=0, FLT_TINY, x)); v_log_f32 path, ample accuracy for f32
    float xe = (v <= 0.0f) ? 1.17549435e-38f : v;
    return __logf(xe);
}

// ---------------------------------------------------------------------------
// Kernel A: st partials = (w^T o) over i-chunks, via V_WMMA_F32_16X16X4_F32.
// grid = (64 j-groups, 8 i-chunks), block = 256 (8 waves). Wave handles one
// 16-column j-tile; A = o[k..k+3] replicated over M, B = 4x16 tile of W.
// The 4 KB o-chunk is staged into LDS by the Tensor Data Mover (1-D D#),
// synchronized with s_wait_tensorcnt.
// ---------------------------------------------------------------------------
__global__ __launch_bounds__(256)
void potts_gemv_wmma(const float* __restrict__ w, const float* __restrict__ o,
                     float* __restrict__ part) {
    __shared__ float olds[1024];
    const int ichunk = blockIdx.y;           // 8 chunks of 1024 rows of W
    const int i0 = ichunk * 1024;

#if __has_builtin(__builtin_amdgcn_tensor_load_to_lds)
    if (threadIdx.x < 32) {                  // one TDM issue per block (wave 0)
        const uint64_t gaddr  = (uint64_t)(const void*)(o + i0);
        const uint32_t ldsoff = (uint32_t)(uintptr_t)(&olds[0]); // LDS = addr[31:0]
        // D# group 0: count=1 | lds_addr | global_addr[56:0] | type=2
        u32x4 g0;
        g0[0] = 1u;                                   // count=1, user desc
        g0[1] = ldsoff;                               // lds_addr (bytes)
        g0[2] = (uint32_t)gaddr;                      // global_addr[31:0]
        g0[3] = ((uint32_t)(gaddr >> 32) & 0x01FFFFFFu) | (2u << 30); // [56:32]|type=2
        // D# group 1: data_size=4B, tensor_dim0=1024, tile_dim0=1024, stride0=1024
        i32x8 g1 = {};
        g1[0] = (int)(2u << 16);                      // data_size=2 -> 4 bytes
        g1[1] = (int)((1024u & 0xFFFFu) << 16);       // tensor_dim0[15:0] @ bits63:48
        g1[2] = (int)(1u << 16);                      // tensor_dim1=1 @ bits95:80
        g1[3] = (int)(1024u << 16);                   // tile_dim0=1024 @ bits127:112
        g1[5] = 1024;                                 // tensor_dim0_stride[31:0]
        i32x4 gz4 = {};                               // groups 2/3 unused (<=2D)
        i32x8 gz8 = {};                               // trailing group (unused)
        __builtin_amdgcn_tensor_load_to_lds(g0, g1, gz4, gz4, gz8, 0);
        __builtin_amdgcn_s_wait_tensorcnt(0);
    }
#else
    for (int t = threadIdx.x; t < 1024; t += 256) olds[t] = o[i0 + t];
#endif
    __syncthreads();

    const int lane = threadIdx.x & 31;
    const int wv   = threadIdx.x >> 5;       // wave in block: 0..7
    const int jt   = blockIdx.x * 8 + wv;    // 16-wide j tile id: 0..511
    const int jc   = jt * 16 + (lane & 15);  // this lane's column (B/D layout)
    const int koff = (lane >> 4) << 1;       // K split: lanes0-15 -> {0,1}, 16-31 -> {2,3}

    const float* wp = w + (size_t)(i0 + koff) * HM + jc;
    const float* al = olds + koff;
    v8f acc = {};
    #pragma unroll 4
    for (int ii = 0; ii < 1024; ii += 4) {
        v2f a, b;
        a.x = al[ii];        // A: K=koff    (M replicated across lanes)
        a.y = al[ii + 1];    // A: K=koff+1
        b.x = wp[0];         // B: row i+koff,   col jc
        b.y = wp[HM];        // B: row i+koff+1, col jc
        __builtin_prefetch(wp + (size_t)32 * HM, 0, 1); // stream-ahead on W
        acc = __builtin_amdgcn_wmma_f32_16x16x4_f32(
                  false, a, false, b, (short)0, acc, false, false);
        wp += (size_t)4 * HM;
    }
    // All D rows identical (A replicated over M); row M=0 lives in acc[0], lanes 0-15.
    if (lane < 16) part[(size_t)ichunk * HM + (size_t)jt * 16 + lane] = acc[0];
}

// ---------------------------------------------------------------------------
// Kernel B: reduce partials, full small-state update incl. softmax over M=8
// (8-lane butterfly shuffles). One thread per element j = h*8+m. grid 32x256.
// ---------------------------------------------------------------------------
__global__ __launch_bounds__(256)
void potts_state(const float* __restrict__ x,  const float* __restrict__ s,
                 const float* __restrict__ a,  const float* __restrict__ zi,
                 const float* __restrict__ zj, const float* __restrict__ p,
                 const float* __restrict__ pi, const float* __restrict__ pj,
                 const float* __restrict__ b,  const float* __restrict__ noise,
                 const float* __restrict__ ga_, const float* __restrict__ gw_,
                 const float* __restrict__ gb_, const float* __restrict__ gs_,
                 const float* __restrict__ k_,  const float* __restrict__ part,
                 float* __restrict__ out) {
    const int j = blockIdx.x * 256 + threadIdx.x;      // 0..8191
    const float ga = ga_[0], gw = gw_[0], gb = gb_[0], gs = gs_[0], k = k_[0];

    float st = b[j];
    #pragma unroll
    for (int c8 = 0; c8 < 8; ++c8) st += part[(size_t)c8 * HM + j];
    st *= gw;

    const float sj = s[j], aj = a[j];
    const float sn = sj + (float)(0.001 / 0.05) *
                     (st - aj + log_eps(x[j]) + gs * noise[j] - sj);

    // softmax over the 8 lanes of this row (groups of 8 aligned in wave32)
    float mx = sn;
    mx = fmaxf(mx, __shfl_xor(mx, 1, 32));
    mx = fmaxf(mx, __shfl_xor(mx, 2, 32));
    mx = fmaxf(mx, __shfl_xor(mx, 4, 32));
    float e  = __expf(sn - mx);
    float se = e;
    se += __shfl_xor(se, 1, 32);
    se += __shfl_xor(se, 2, 32);
    se += __shfl_xor(se, 4, 32);
    const float on = e / se;

    const float an  = aj    + (float)(0.001 / 2.7)  * (ga * on - aj);
    const float zin = zi[j] + (float)(0.001 / 0.24) * (on - zi[j]);
    const float zjn = zj[j] + (float)(0.001 / 0.24) * (on - zj[j]);
    const float c   = k * (float)(0.001 / 10.0);
    const float pin = pi[j] + c * (zin - pi[j]);
    const float pjn = pj[j] + c * (zjn - pj[j]);
    const float bn  = gb * log_eps(pjn);

    out[OFF_O  + j] = on;
    out[OFF_S  + j] = sn;
    out[OFF_A  + j] = an;
    out[OFF_ZI + j] = zin;
    out[OFF_ZJ + j] = zjn;
    out[OFF_PI + j] = pin;
    out[OFF_PJ + j] = pjn;
    out[OFF_B  + j] = bn;
    if (j == 0) out[OFF_P] = p[0] + c * (1.0f - p[0]);
}

// ---------------------------------------------------------------------------
// Kernel C: fused pij_new + w_new over 8192^2 (768 MB of traffic). One block
// per row r: zi[r]/pi[r] are uniform (scalar loads); pij streamed with NT
// b128 loads; both outputs stream out with non-temporal stores.
// ---------------------------------------------------------------------------
__global__ __launch_bounds__(256)
void potts_joint(const float* __restrict__ pij, const float* __restrict__ k_,
                 float* __restrict__ out) {
    const int   r     = blockIdx.x;
    const float c     = k_[0] * (float)(0.001 / 10.0);
    const float p_new = out[OFF_P];
    const float zi_r  = out[OFF_ZI + r];
    const float pi_r  = out[OFF_PI + r];

    const size_t rowoff = (size_t)r * HM;
    const float* __restrict__ zjv  = out + OFF_ZJ;
    const float* __restrict__ pjv  = out + OFF_PJ;
    const float* __restrict__ prow = pij + rowoff;
    float* __restrict__ pout = out + OFF_PIJ + rowoff;   // odd base -> scalar stores
    float* __restrict__ wout = out + OFF_W   + rowoff;

    for (int base = threadIdx.x * 4; base < HM; base += 256 * 4) {
        v4f pv = __builtin_nontemporal_load((const v4f*)(prow + base));
        #pragma unroll
        for (int u = 0; u < 4; ++u) {
            const int   cc  = base + u;
            const float pn  = pv[u] + c * (zi_r * zjv[cc] - pv[u]);
            const float q   = __fdividef(p_new * pn, pi_r * pjv[cc]);
            const float wn  = log_eps(q);
            __builtin_nontemporal_store(pn, pout + cc);
            __builtin_nontemporal_store(wn, wout + cc);
        }
    }
}

// ---------------------------------------------------------------------------
extern "C" void kernel_launch(void* const* d_in, const int* in_sizes, int n_in,
                              void* d_out, int out_size, void* d_ws, size_t ws_size,
                              hipStream_t stream) {
    const float* x     = (const float*)d_in[0];
    const float* s     = (const float*)d_in[1];
    const float* o     = (const float*)d_in[2];
    const float* a     = (const float*)d_in[3];
    const float* zi    = (const float*)d_in[4];
    const float* zj    = (const float*)d_in[5];
    const float* p     = (const float*)d_in[6];
    const float* pi    = (const float*)d_in[7];
    const float* pj    = (const float*)d_in[8];
    const float* pij   = (const float*)d_in[9];
    const float* b     = (const float*)d_in[10];
    const float* w     = (const float*)d_in[11];
    const float* noise = (const float*)d_in[12];
    const float* ga    = (const float*)d_in[13];
    const float* gw    = (const float*)d_in[14];
    const float* gb    = (const float*)d_in[15];
    const float* gs    = (const float*)d_in[16];
    const float* kk    = (const float*)d_in[17];
    float* out  = (float*)d_out;
    float* part = (float*)d_ws;   // 8 chunks * 8192 floats = 256 KB partials

    dim3 gA(64, 8);
    potts_gemv_wmma<<<gA, 256, 0, stream>>>(w, o, part);
    potts_state<<<32, 256, 0, stream>>>(x, s, a, zi, zj, p, pi, pj, b, noise,
                                        ga, gw, gb, gs, kk, part, out);
    potts_joint<<<HM, 256, 0, stream>>>(pij, kk, out);
}